// TokenSelector_18348100288647
// MI455X (gfx1250) — compile-verified
//
#include <hip/hip_runtime.h>

// Problem constants (match the reference).
#define BROWS 2048
#define NCOLS 16384
#define KSEL  256
#define NTHR  512
#define NWAVE (NTHR / 32)        // wave32 on gfx1250
#define VALS  (NCOLS / NTHR)     // 32 elements per thread, register resident

// 16-byte int vector type matching the async-copy builtin's pointer pointee.
typedef __attribute__((__vector_size__(16))) int v4i_t;

// Order-preserving map: f32 -> u32 (ascending).
__device__ __forceinline__ unsigned ordmap(float x) {
  unsigned u = __float_as_uint(x);
  return (u & 0x80000000u) ? ~u : (u | 0x80000000u);
}

__device__ __forceinline__ float waveSum(float v) {
#pragma unroll
  for (int o = 16; o > 0; o >>= 1) v += __shfl_xor(v, o, 32);
  return v;
}
__device__ __forceinline__ float waveMax(float v) {
#pragma unroll
  for (int o = 16; o > 0; o >>= 1) v = fmaxf(v, __shfl_xor(v, o, 32));
  return v;
}

// Block-wide sum, result broadcast to all threads.
__device__ __forceinline__ float blockSum(float v, float* red, int tid) {
  v = waveSum(v);
  const int wave = tid >> 5, lane = tid & 31;
  if (lane == 0) red[wave] = v;
  __syncthreads();
  if (wave == 0) {
    float y = (lane < NWAVE) ? red[lane] : 0.0f;
    y = waveSum(y);
    if (lane == 0) red[0] = y;
  }
  __syncthreads();
  const float r = red[0];
  __syncthreads();
  return r;
}

// Fused block-wide sum of two values (halves barrier cost in the Newton loop).
__device__ __forceinline__ void blockSum2(float& a, float& b, float* red, int tid) {
#pragma unroll
  for (int o = 16; o > 0; o >>= 1) {
    a += __shfl_xor(a, o, 32);
    b += __shfl_xor(b, o, 32);
  }
  const int wave = tid >> 5, lane = tid & 31;
  if (lane == 0) { red[2 * wave] = a; red[2 * wave + 1] = b; }
  __syncthreads();
  if (wave == 0) {
    float x = (lane < NWAVE) ? red[2 * lane] : 0.0f;
    float y = (lane < NWAVE) ? red[2 * lane + 1] : 0.0f;
#pragma unroll
    for (int o = 16; o > 0; o >>= 1) {
      x += __shfl_xor(x, o, 32);
      y += __shfl_xor(y, o, 32);
    }
    if (lane == 0) { red[0] = x; red[1] = y; }
  }
  __syncthreads();
  a = red[0];
  b = red[1];
  __syncthreads();
}

__device__ __forceinline__ float blockMax(float v, float* red, int tid) {
  v = waveMax(v);
  const int wave = tid >> 5, lane = tid & 31;
  if (lane == 0) red[wave] = v;
  __syncthreads();
  if (wave == 0) {
    float y = (lane < NWAVE) ? red[lane] : -3.402823466e38f;
    y = waveMax(y);
    if (lane == 0) red[0] = y;
  }
  __syncthreads();
  const float r = red[0];
  __syncthreads();
  return r;
}

// One workgroup per row.  Row staged once into LDS via the CDNA5 async copy
// path, then pulled into VGPRs (32/thread); every subsequent sweep is pure
// VALU.  tau via 8 Newton iterations on g(tau)=sum(max(z-tau,0)^2)-1 (convex,
// monotone from tau0=zmax-1); exact K-th largest via 4-round 256-bin radix
// select with LDS atomics.  HBM traffic: 1 read + 1 write per element.
__global__ __launch_bounds__(NTHR)
void entmax15_topk_kernel(const float* __restrict__ logits,
                          float* __restrict__ out_w,
                          float* __restrict__ out_cnt) {
  __shared__ alignas(16) float zbuf[NCOLS];   // 64 KB stage buffer
  __shared__ float red[2 * NWAVE];
  __shared__ unsigned hist[256];
  __shared__ unsigned suffix[256];
  __shared__ int selBin;
  __shared__ unsigned sh_prefix;
  __shared__ int sh_krem;

  const int row = blockIdx.x;
  const int tid = threadIdx.x;
  const float* __restrict__ src = logits + (size_t)row * NCOLS;

  // ---- Stage the row into LDS (async global->LDS, CDNA5 path) -----------
#if defined(__gfx1250__) && __has_builtin(__builtin_amdgcn_global_load_async_to_lds_b128)
  {
    const char* g = (const char*)src;
    char* l = (char*)zbuf;
#pragma unroll
    for (int it = 0; it < (NCOLS * 4) / (NTHR * 16); ++it) {   // 8 x b128 per thread
      const int off = (tid + it * NTHR) * 16;
      __builtin_amdgcn_global_load_async_to_lds_b128(
          (__attribute__((address_space(1))) v4i_t*)(g + off),
          (__attribute__((address_space(3))) v4i_t*)(l + off),
          0, 0);
    }
#if __has_builtin(__builtin_amdgcn_s_wait_asynccnt)
    __builtin_amdgcn_s_wait_asynccnt(0);
#else
    asm volatile("s_wait_asynccnt 0" ::: "memory");
#endif
  }
#else
  for (int i = tid * 4; i < NCOLS; i += NTHR * 4) {
    const float4 v = *(const float4*)(src + i);
    *(float4*)(zbuf + i) = v;
  }
#endif
  __syncthreads();

  // ---- LDS -> VGPRs, bank-conflict-free stride-NTHR layout --------------
  float rv[VALS];
#pragma unroll
  for (int j = 0; j < VALS; ++j) rv[j] = zbuf[tid + j * NTHR];

  // ---- Row max (z = logits * 0.5, TEMPERATURE == 1) ---------------------
  float m = -3.402823466e38f;
#pragma unroll
  for (int j = 0; j < VALS; ++j) m = fmaxf(m, rv[j]);
  m = blockMax(m, red, tid);
  const float zmax = 0.5f * m;

  // ---- Newton for tau: sum(max(z - tau, 0)^2) == 1 ----------------------
  // g convex & decreasing; g(zmax-1) >= 1 so iterates rise monotonically to
  // the root with quadratic convergence.
  float tau = zmax - 1.0f;
#pragma unroll 1
  for (int it = 0; it < 8; ++it) {
    float s = 0.0f, t = 0.0f;
#pragma unroll
    for (int j = 0; j < VALS; ++j) {
      const float d = fmaxf(fmaf(0.5f, rv[j], -tau), 0.0f);
      s = fmaf(d, d, s);
      t += d;
    }
    blockSum2(s, t, red, tid);            // broadcast -> tau stays uniform
    tau += (s - 1.0f) / fmaxf(2.0f * t, 1e-12f);
  }

  // ---- Support size -----------------------------------------------------
  float c = 0.0f;
#pragma unroll
  for (int j = 0; j < VALS; ++j)
    c += (0.5f * rv[j] > tau) ? 1.0f : 0.0f;
  c = blockSum(c, red, tid);
  const bool needK = (c > (float)KSEL);   // uniform across the block

  // ---- Exact K-th largest via 4-round 8-bit radix select ----------------
  unsigned ucut = 0u;
  if (needK) {
    if (tid == 0) { sh_prefix = 0u; sh_krem = KSEL; }
    __syncthreads();
#pragma unroll 1
    for (int r = 0; r < 4; ++r) {
      const int shift = 24 - 8 * r;
      if (tid < 256) hist[tid] = 0u;
      if (tid == 0) selBin = 0;
      __syncthreads();
      const unsigned pref = sh_prefix;
      const unsigned hmask = (r == 0) ? 0u : (0xFFFFFFFFu << (shift + 8));
#pragma unroll
      for (int j = 0; j < VALS; ++j) {
        const unsigned mp = ordmap(rv[j]);
        if ((mp & hmask) == pref)
          atomicAdd(&hist[(mp >> shift) & 255u], 1u);
      }
      __syncthreads();
      // suffix[b] = sum_{d >= b} hist[d]  (Hillis-Steele over 256 bins)
      if (tid < 256) suffix[tid] = hist[tid];
      __syncthreads();
#pragma unroll 1
      for (int o = 1; o < 256; o <<= 1) {
        unsigned add = 0u;
        if (tid < 256 && tid + o < 256) add = suffix[tid + o];
        __syncthreads();
        if (tid < 256) suffix[tid] += add;
        __syncthreads();
      }
      // largest bin b with suffix[b] >= krem (suffix is non-increasing in b)
      const int krem = sh_krem;
      if (tid < 256 && (int)suffix[tid] >= krem) atomicMax(&selBin, tid);
      __syncthreads();
      if (tid == 0) {
        const int b = selBin;
        const unsigned above = (b < 255) ? suffix[b + 1] : 0u;
        sh_krem = krem - (int)above;
        sh_prefix = pref | ((unsigned)b << shift);
      }
      __syncthreads();
    }
    ucut = sh_prefix;   // map value of the exact K-th largest element
  }

  // ---- Sum of selected weights ------------------------------------------
  float S = 0.0f;
#pragma unroll
  for (int j = 0; j < VALS; ++j) {
    const float zi = 0.5f * rv[j];
    const bool sel = needK ? (ordmap(rv[j]) >= ucut) : (zi > tau);
    if (sel) {
      const float d = fmaxf(zi - tau, 0.0f);
      S = fmaf(d, d, S);
    }
  }
  S = blockSum(S, red, tid);
  const float inv = 1.0f / (S + 1e-8f);

  // ---- Write normalized weights (coalesced) + count ---------------------
  float ns = 0.0f;
  float* __restrict__ dst = out_w + (size_t)row * NCOLS;
#pragma unroll
  for (int j = 0; j < VALS; ++j) {
    const float zi = 0.5f * rv[j];
    const bool sel = needK ? (ordmap(rv[j]) >= ucut) : (zi > tau);
    float w = 0.0f;
    if (sel) {
      const float d = fmaxf(zi - tau, 0.0f);
      w = d * d * inv;
    }
    dst[tid + j * NTHR] = w;
    ns += (w > 1e-6f) ? 1.0f : 0.0f;
  }
  ns = blockSum(ns, red, tid);
  if (tid == 0) out_cnt[row] = ns;
}

extern "C" void kernel_launch(void* const* d_in, const int* in_sizes, int n_in,
                              void* d_out, int out_size, void* d_ws, size_t ws_size,
                              hipStream_t stream) {
  (void)in_sizes; (void)n_in; (void)out_size; (void)d_ws; (void)ws_size;
  const float* logits = (const float*)d_in[0];
  float* out_w   = (float*)d_out;                          // B*N weights
  float* out_cnt = out_w + (size_t)BROWS * NCOLS;          // B counts (as float)
  entmax15_topk_kernel<<<BROWS, NTHR, 0, stream>>>(logits, out_w, out_cnt);
}